// ForwardSamplerMultiscale3d_72954314490428
// MI455X (gfx1250) — compile-verified
//
#include <hip/hip_runtime.h>
#include <cstdint>
#include <cstddef>

#define OH 512
#define OW 512
#define MAX_RAY 100

// ---------------------------------------------------------------------------
// CDNA5 async global->LDS copy (gfx1250). Per-lane b64 copy, GVS addressing:
//   mem_addr = SADDR(64) + VADDR(32) + inst_offset ; dst VGPR = LDS byte addr.
// Tracked with ASYNCcnt; each wave drains its own copies with s_wait_asynccnt,
// then the workgroup barrier makes every wave's LDS slots visible to all.
// ---------------------------------------------------------------------------
__device__ __forceinline__ uint32_t lds_byte_addr(const void* p) {
    // Generic (flat) LDS addresses carry the wave-relative LDS offset in the
    // low 32 bits (ISA 10.2 aperture mapping) — truncate to get the DS address.
    return (uint32_t)(size_t)p;
}

__device__ __forceinline__ void async_load_b64(void* lds_dst, const void* base,
                                               uint32_t byte_off) {
    uint32_t lds = lds_byte_addr(lds_dst);
    asm volatile("global_load_async_to_lds_b64 %0, %1, %2"
                 :
                 : "v"(lds), "v"(byte_off), "s"((uint64_t)base)
                 : "memory");
}

__device__ __forceinline__ void wait_async_all() {
    asm volatile("s_wait_asynccnt 0x0" ::: "memory");
}

// ---------------------------------------------------------------------------
// Zero fill (b128 stores for the 142 MB output; scalar tail kernel)
// ---------------------------------------------------------------------------
__global__ void zero_f32v4(float4* p, int n4) {
    float4 z = make_float4(0.f, 0.f, 0.f, 0.f);
    for (int i = blockIdx.x * blockDim.x + threadIdx.x; i < n4;
         i += gridDim.x * blockDim.x)
        p[i] = z;
}

__global__ void zero_f32(float* p, int n) {
    for (int i = blockIdx.x * blockDim.x + threadIdx.x; i < n;
         i += gridDim.x * blockDim.x)
        p[i] = 0.0f;
}

__global__ void zero_i32(int* p, int n) {
    for (int i = blockIdx.x * blockDim.x + threadIdx.x; i < n;
         i += gridDim.x * blockDim.x)
        p[i] = 0;
}

// ---------------------------------------------------------------------------
// Cell id helpers: c = floor(clip(coord*512, 0, 511)) then >> shift
// ---------------------------------------------------------------------------
__device__ __forceinline__ int cell_of(float cx, float cy, int shift, int w) {
    float fx = fminf(fmaxf(cx * (float)OH, 0.0f), (float)(OH - 1));
    float fy = fminf(fmaxf(cy * (float)OW, 0.0f), (float)(OW - 1));
    int ci = ((int)fx) >> shift;   // truncation == floor (non-negative)
    int cj = ((int)fy) >> shift;
    return ci * w + cj;
}

__global__ void count_points(const float2* __restrict__ coords, int n,
                             int shift, int w, int* __restrict__ cnt) {
    for (int p = blockIdx.x * blockDim.x + threadIdx.x; p < n;
         p += gridDim.x * blockDim.x) {
        float2 xy = coords[p];
        int cell = cell_of(xy.x, xy.y, shift, w);
        atomicAdd(&cnt[cell], 1);
    }
}

// ---------------------------------------------------------------------------
// Two-level exclusive scan of packed u64 { occupancy_flag:hi32 | count:lo32 }.
// Low 32 -> span start offsets; high 32 -> compacted occupied-cell column id.
// ---------------------------------------------------------------------------
__global__ void scan_block(const int* __restrict__ cnt,
                           unsigned long long* __restrict__ packedEx,
                           unsigned long long* __restrict__ blockSums, int n) {
    __shared__ unsigned long long s[256];
    int gid = blockIdx.x * 256 + threadIdx.x;
    int c = (gid < n) ? cnt[gid] : 0;
    unsigned long long v = (unsigned long long)(unsigned)c |
                           ((unsigned long long)(c > 0) << 32);
    s[threadIdx.x] = v;
    __syncthreads();
    for (int off = 1; off < 256; off <<= 1) {
        unsigned long long t =
            (threadIdx.x >= (unsigned)off) ? s[threadIdx.x - off] : 0ull;
        __syncthreads();
        s[threadIdx.x] += t;
        __syncthreads();
    }
    if (gid < n) packedEx[gid] = s[threadIdx.x] - v;   // exclusive within block
    if (threadIdx.x == 255) blockSums[blockIdx.x] = s[255];
}

__global__ void scan_sums(unsigned long long* __restrict__ bs, int nb) {
    __shared__ unsigned long long s[1024];
    int t = threadIdx.x;
    unsigned long long v = (t < nb) ? bs[t] : 0ull;
    s[t] = v;
    __syncthreads();
    for (int off = 1; off < 1024; off <<= 1) {
        unsigned long long x = (t >= off) ? s[t - off] : 0ull;
        __syncthreads();
        s[t] += x;
        __syncthreads();
    }
    if (t < nb) bs[t] = s[t] - v;                      // exclusive
}

__global__ void scan_finalize(const unsigned long long* __restrict__ packedEx,
                              const unsigned long long* __restrict__ bs,
                              int* __restrict__ start, int* __restrict__ col,
                              int* __restrict__ cursor, int n) {
    int gid = blockIdx.x * blockDim.x + threadIdx.x;
    if (gid < n) {
        unsigned long long v = packedEx[gid] + bs[gid >> 8];
        int st = (int)(unsigned)v;
        start[gid] = st;
        col[gid] = (int)(v >> 32);
        cursor[gid] = st;
    }
}

// ---------------------------------------------------------------------------
// Counting-sort points into contiguous per-cell spans of packed 8-byte
// records { depth:f32, orig_idx:u32 } (one global_store_b64 per point).
// Order within a span is irrelevant: ranking below is symmetric pairwise.
// ---------------------------------------------------------------------------
__global__ void fill_points(const float2* __restrict__ coords,
                            const float* __restrict__ depth, int n, int shift,
                            int w, int* __restrict__ cursor,
                            uint2* __restrict__ pdi) {
    for (int p = blockIdx.x * blockDim.x + threadIdx.x; p < n;
         p += gridDim.x * blockDim.x) {
        float2 xy = coords[p];
        int cell = cell_of(xy.x, xy.y, shift, w);
        int slot = atomicAdd(&cursor[cell], 1);
        pdi[slot] = make_uint2(__float_as_uint(depth[p]), (unsigned)p);
    }
}

// ---------------------------------------------------------------------------
// One 256-thread block per cell. Stage span tiles of (depth,idx) pairs into
// LDS via async b64 global->LDS copies; rank_i = #{ j : (d_j,idx_j) <
// (d_i,idx_i) } (matches stable argsort-by-depth then stable sort-by-cell in
// the reference); scatter payloads of ranks < MAX_RAY into batch[rank*HW+col].
// ---------------------------------------------------------------------------
__global__ void rank_scatter(const uint2* __restrict__ pdi,
                             const float* __restrict__ pm,
                             const int* __restrict__ start,
                             const int* __restrict__ cnt,
                             const int* __restrict__ col,
                             float* __restrict__ out_batch,
                             float* __restrict__ out_len,
                             float* __restrict__ out_canvas, int w, int HW) {
    int cell = blockIdx.x;
    int c = cnt[cell];
    if (c == 0) return;                       // uniform exit: whole block
    int st = start[cell];
    int cc = col[cell];
    if (threadIdx.x == 0) {
        out_len[cc] = (float)(c < MAX_RAY ? c : MAX_RAY);
        out_canvas[2 * cc] = (float)(cell / w);
        out_canvas[2 * cc + 1] = (float)(cell % w);
    }

    __shared__ uint2 sp[256];                 // 8-byte aligned pair tile

    for (int ibase = 0; ibase < c; ibase += 256) {
        int i = ibase + (int)threadIdx.x;
        bool act = (i < c);
        int ic = act ? i : (c - 1);
        uint2 rec = pdi[st + ic];
        float di = __uint_as_float(rec.x);
        int ii = (int)rec.y;
        if (act) __builtin_prefetch(&pm[ii], 0, 0);  // global_prefetch
        int rank = 0;
        for (int tb = 0; tb < c; tb += 256) {
            int j = tb + (int)threadIdx.x;
            int jc = (j < c) ? j : (c - 1);          // clamp: no divergent asm
            __syncthreads();                          // protect prior tile reads
            async_load_b64(&sp[threadIdx.x], pdi, (uint32_t)(st + jc) * 8u);
            wait_async_all();                         // this wave's slots done
            __syncthreads();                          // => all waves' slots done
            int tile = c - tb;
            if (tile > 256) tile = 256;
            if (act) {
                for (int t = 0; t < tile; ++t) {
                    uint2 r = sp[t];                  // one ds_load_b64
                    float dj = __uint_as_float(r.x);
                    int ij = (int)r.y;
                    rank += (dj < di) || (dj == di && ij < ii);
                }
            }
        }
        if (act && rank < MAX_RAY) out_batch[rank * HW + cc] = pm[ii];
    }
}

// ---------------------------------------------------------------------------
// Launcher
// ---------------------------------------------------------------------------
extern "C" void kernel_launch(void* const* d_in, const int* in_sizes, int n_in,
                              void* d_out, int out_size, void* d_ws,
                              size_t ws_size, hipStream_t stream) {
    const float2* coords = (const float2*)d_in[0];
    const float* depth = (const float*)d_in[1];
    const float* pm = (const float*)d_in[2];
    const int n = in_sizes[1];               // N points
    float* out = (float*)d_out;

    // Workspace bump allocator (buffers reused across the 3 sequential scales).
    char* ws = (char*)d_ws;
    size_t off = 0;
    auto alloc = [&](size_t bytes) -> void* {
        off = (off + 15) & ~(size_t)15;
        void* p = ws + off;
        off += bytes;
        return p;
    };
    const int HW1 = (OH) * (OW);             // 262144, largest scale
    int* cnt = (int*)alloc((size_t)HW1 * 4);
    int* start = (int*)alloc((size_t)HW1 * 4);
    int* col = (int*)alloc((size_t)HW1 * 4);
    int* cursor = (int*)alloc((size_t)HW1 * 4);
    unsigned long long* packedEx = (unsigned long long*)alloc((size_t)HW1 * 8);
    unsigned long long* bsums = (unsigned long long*)alloc(1024 * 8);
    uint2* pdi = (uint2*)alloc((size_t)n * 8);

    // Zero all outputs (padded columns / unused ranks must read 0).
    {
        int n4 = out_size >> 2;
        int rem = out_size & 3;
        if (n4 > 0) zero_f32v4<<<4096, 256, 0, stream>>>((float4*)out, n4);
        if (rem > 0)
            zero_f32<<<1, 256, 0, stream>>>(out + (size_t)n4 * 4, rem);
    }

    struct Sc {
        int shift, w, HW;
        size_t ob, ol, oc;                   // float offsets into d_out
    };
    const Sc sc[3] = {
        {0, 512, 262144, 0u, 34406400u, 34750464u},
        {1, 256, 65536, 26214400u, 34668544u, 35274752u},
        {2, 128, 16384, 32768000u, 34734080u, 35405824u},
    };

    for (int k = 0; k < 3; ++k) {
        const int HW = sc[k].HW, w = sc[k].w, shift = sc[k].shift;
        const int nb = HW / 256;
        zero_i32<<<(HW + 255) / 256, 256, 0, stream>>>(cnt, HW);
        count_points<<<2048, 256, 0, stream>>>(coords, n, shift, w, cnt);
        scan_block<<<nb, 256, 0, stream>>>(cnt, packedEx, bsums, HW);
        scan_sums<<<1, 1024, 0, stream>>>(bsums, nb);
        scan_finalize<<<nb, 256, 0, stream>>>(packedEx, bsums, start, col,
                                              cursor, HW);
        fill_points<<<2048, 256, 0, stream>>>(coords, depth, n, shift, w,
                                              cursor, pdi);
        rank_scatter<<<HW, 256, 0, stream>>>(pdi, pm, start, cnt, col,
                                             out + sc[k].ob, out + sc[k].ol,
                                             out + sc[k].oc, w, HW);
    }
}